// QuantumLayer_32023276159181
// MI455X (gfx1250) — compile-verified
//
#include <hip/hip_runtime.h>

// out[b] = cos(x[b,0] + w[0])
//
// Derivation (exact): RY layers on |0..0> compose additively per qubit ->
// product state with angles (x[b,q]+w[q])/2. The CNOT chain (targets 1..7)
// never flips qubit 0, so <Z0> is invariant under it:
//   out = cos^2((x0+w0)/2) - sin^2((x0+w0)/2) = cos(x0 + w0).
// The kernel is HBM-bound: ~8MB strided read of x (every 128B line of x
// contains 4 needed words) + 1MB write => ~0.4us at 23.3 TB/s.
//
// Data path: stage the needed words through LDS with the CDNA5 async
// global->LDS DMA (ASYNCcnt / s_wait_asynccnt), then v_cos_f32 + store.
// __cosf (native, v_cos_f32) is a few-ulp exact for |arg| <~ 8 (x ~ N(0,1)),
// keeping the VALU side negligible vs. the memory bound.

#define TPB 256
#define EPT 4                 // elements (outputs) per thread
#define TILE (TPB * EPT)      // 1024 outputs per block

typedef __attribute__((address_space(1))) int gint_t;   // global int
typedef __attribute__((address_space(3))) int lint_t;   // LDS int

__device__ __forceinline__ void async_g2lds_b32(const float* gptr, float* lptr) {
#if __has_builtin(__builtin_amdgcn_global_load_async_to_lds_b32)
    __builtin_amdgcn_global_load_async_to_lds_b32(
        (gint_t*)(void*)gptr, (lint_t*)(void*)lptr, /*offset=*/0, /*cpol=*/0);
#else
    unsigned lds_off = (unsigned)(unsigned long)((lint_t*)(void*)lptr);
    asm volatile("global_load_async_to_lds_b32 %0, %1, off"
                 :
                 : "v"(lds_off), "v"(gptr)
                 : "memory");
#endif
}

__device__ __forceinline__ void wait_asynccnt0() {
#if __has_builtin(__builtin_amdgcn_s_wait_asynccnt)
    __builtin_amdgcn_s_wait_asynccnt(0);
#else
    asm volatile("s_wait_asynccnt 0" ::: "memory");
#endif
}

__global__ __launch_bounds__(TPB) void
QuantumLayer_kernel(const float* __restrict__ x,   // [B, 8]
                    const float* __restrict__ w,   // [8]
                    float* __restrict__ out,       // [B]
                    int B) {
    __shared__ float tile[TILE];

    const int tid  = threadIdx.x;
    const int base = blockIdx.x * TILE;

    // Phase 1: async DMA of x[(base+idx)*8 + 0] into LDS slot idx.
    // Lane-consecutive idx -> 32B lane stride -> 1KB per wave-issue, fully
    // covering the cache lines we are forced to pull anyway.
#pragma unroll
    for (int k = 0; k < EPT; ++k) {
        int idx = tid + k * TPB;
        int b   = base + idx;
        int bc  = (b < B) ? b : (B - 1);       // clamp (B is a multiple of TILE anyway)
        async_g2lds_b32(x + (size_t)bc * 8, &tile[idx]);
    }
    wait_asynccnt0();   // this wave's DMA complete; each thread reads only its own slots

    const float w0 = w[0];                      // scalar (SMEM) load, broadcast

    // Phase 2: out[b] = cos(x[b,0] + w0), native v_cos_f32 path
#pragma unroll
    for (int k = 0; k < EPT; ++k) {
        int idx = tid + k * TPB;
        int b   = base + idx;
        if (b < B) {
            out[b] = __cosf(tile[idx] + w0);
        }
    }
}

extern "C" void kernel_launch(void* const* d_in, const int* in_sizes, int n_in,
                              void* d_out, int out_size, void* d_ws, size_t ws_size,
                              hipStream_t stream) {
    const float* x = (const float*)d_in[0];   // [B, 8] float32
    const float* w = (const float*)d_in[1];   // [8] float32
    float* out     = (float*)d_out;           // [B] float32

    const int B    = in_sizes[0] / 8;         // N_QUBITS = 8
    const int grid = (B + TILE - 1) / TILE;   // 256 blocks for B = 262144

    QuantumLayer_kernel<<<grid, TPB, 0, stream>>>(x, w, out, B);
}